// ARModel_26757646254132
// MI455X (gfx1250) — compile-verified
//
#include <hip/hip_runtime.h>

// ARModel on MI455X (gfx1250, wave32).
// Pipeline (all on `stream`):
//   1) preprocess_kernel : h = relu([x,emb]@W_pre1+b)@W_pre2+b  -> h2[t][l] (float2, transposed, coalesced for RNN)
//   2) ytranspose_kernel : y[l][t] -> yT[t][l]                   (coalesced RNN reads)
//   3) rnn_kernel        : 1 lane per sequence, serial scan; packed V_PK_FMA_F32 math,
//                          shallow reduction tree on the recurrence, 4x v_tanh_f32 in one
//                          asm block (single hazard v_nop), global_prefetch for next steps.
//   4) head_kernel       : per 16-row tile: V_WMMA_F32_16X16X4_F32 for s@W_d1 (K=4 exact),
//                          relu + bias, tiny 6x2 epilogue via LDS, coalesced out[L,4095,2]
//
// Workspace: h2 32MB @0, yT 8MB @32MB, sT 64MB @40MB (~109MB total, fits in 192MB L2).

#define L_SEQ   1024
#define T_LEN   4096
#define F_IN    8
#define NY_     2048
#define EMB_    4
#define D1_     6
#define T_OUT   (T_LEN - 1)   // 4095

typedef float v2f __attribute__((ext_vector_type(2)));
typedef float v8f __attribute__((ext_vector_type(8)));

// Four independent hardware tanh ops issued back-to-back: each covers the
// previous one's trans-op hazard; one trailing v_nop covers the last.
__device__ __forceinline__ void tanh4(float n0, float n1, float n2, float n3,
                                      float& s0, float& s1, float& s2, float& s3) {
    float r0, r1, r2, r3;
    asm volatile(
        "v_tanh_f32 %0, %4\n\t"
        "v_tanh_f32 %1, %5\n\t"
        "v_tanh_f32 %2, %6\n\t"
        "v_tanh_f32 %3, %7\n\t"
        "v_nop"
        : "=&v"(r0), "=&v"(r1), "=&v"(r2), "=&v"(r3)
        : "v"(n0), "v"(n1), "v"(n2), "v"(n3));
    s0 = r0; s1 = r1; s2 = r2; s3 = r3;
}

// ---------------------------------------------------------------------------
// 1) Preprocess: per (l,t) compute 2-channel h, store transposed h2[t*L + l].
//    Tile 16 l x 64 t per 256-thread block; LDS transpose for coalesced writes.
// ---------------------------------------------------------------------------
__global__ __launch_bounds__(256) void preprocess_kernel(
    const float* __restrict__ x, const float* __restrict__ embed,
    const float* __restrict__ W1, const float* __restrict__ b1,
    const float* __restrict__ W2, const float* __restrict__ b2,
    float2* __restrict__ h2)
{
    __shared__ float2 tile[16][65];
    const int t0 = blockIdx.x * 64;
    const int l0 = blockIdx.y * 16;

    // uniform weights -> scalar regs
    float rw1[12][4], rb1[4], rw2[4][2], rb2[2];
    #pragma unroll
    for (int k = 0; k < 12; ++k)
        #pragma unroll
        for (int j = 0; j < 4; ++j) rw1[k][j] = W1[k * 4 + j];
    #pragma unroll
    for (int j = 0; j < 4; ++j) rb1[j] = b1[j];
    #pragma unroll
    for (int j = 0; j < 4; ++j) { rw2[j][0] = W2[j * 2]; rw2[j][1] = W2[j * 2 + 1]; }
    rb2[0] = b2[0]; rb2[1] = b2[1];

    #pragma unroll
    for (int k = 0; k < 4; ++k) {
        int e  = k * 256 + threadIdx.x;
        int li = e >> 6, ti = e & 63;
        int l = l0 + li, t = t0 + ti;
        const float* xp = x + ((size_t)l * T_LEN + t) * F_IN;
        float4 xa = *(const float4*)xp;
        float4 xb = *(const float4*)(xp + 4);
        float e0 = embed[l * EMB_ + 0], e1 = embed[l * EMB_ + 1];
        float e2 = embed[l * EMB_ + 2], e3 = embed[l * EMB_ + 3];
        float in[12] = { xa.x, xa.y, xa.z, xa.w, xb.x, xb.y, xb.z, xb.w, e0, e1, e2, e3 };
        float z[4];
        #pragma unroll
        for (int j = 0; j < 4; ++j) {
            float s = rb1[j];
            #pragma unroll
            for (int q = 0; q < 12; ++q) s += in[q] * rw1[q][j];
            z[j] = s > 0.f ? s : 0.f;
        }
        float o0 = rb2[0], o1 = rb2[1];
        #pragma unroll
        for (int j = 0; j < 4; ++j) { o0 += z[j] * rw2[j][0]; o1 += z[j] * rw2[j][1]; }
        tile[li][ti] = make_float2(o0, o1);
    }
    __syncthreads();
    #pragma unroll
    for (int k = 0; k < 4; ++k) {
        int e  = k * 256 + threadIdx.x;
        int ti = e >> 4, li = e & 15;
        h2[(size_t)(t0 + ti) * L_SEQ + (l0 + li)] = tile[li][ti];
    }
}

// ---------------------------------------------------------------------------
// 2) y[l][t] -> yT[t][l], 32x32 LDS tile transpose
// ---------------------------------------------------------------------------
__global__ __launch_bounds__(256) void ytranspose_kernel(
    const float* __restrict__ y, float* __restrict__ yT)
{
    __shared__ float tile[32][33];
    const int t0 = blockIdx.x * 32;
    const int l0 = blockIdx.y * 32;
    #pragma unroll
    for (int k = 0; k < 4; ++k) {
        int e = k * 256 + threadIdx.x;
        int li = e >> 5, ti = e & 31;
        tile[li][ti] = y[(size_t)(l0 + li) * NY_ + (t0 + ti)];
    }
    __syncthreads();
    #pragma unroll
    for (int k = 0; k < 4; ++k) {
        int e = k * 256 + threadIdx.x;
        int ti = e >> 5, li = e & 31;
        yT[(size_t)(t0 + ti) * L_SEQ + (l0 + li)] = tile[li][ti];
    }
}

// ---------------------------------------------------------------------------
// 3) Serial RNN scan: one lane per sequence; 32 single-wave workgroups spread
//    across WGPs. Hidden state as two packed f32 pairs (A = units 0,1 and
//    B = units 2,3) so the recurrence runs on V_PK_FMA_F32 with an explicit
//    3-level tree; input-only terms are computed off the recurrence chain.
// ---------------------------------------------------------------------------
__global__ __launch_bounds__(32) void rnn_kernel(
    const float* __restrict__ yT, const float2* __restrict__ h2,
    const float* __restrict__ Wi1, const float* __restrict__ Wh1, const float* __restrict__ b1,
    const float* __restrict__ Wi2, const float* __restrict__ Wh2, const float* __restrict__ b2,
    float4* __restrict__ sT)
{
    const int l = blockIdx.x * 32 + threadIdx.x;

    // Weight row k, packed as units {0,1} (A) and {2,3} (B).
    v2f wi1A[3], wi1B[3], wh1A[4], wh1B[4], wi2A[2], wi2B[2], wh2A[4], wh2B[4], b1A, b1B, b2A, b2B;
    #pragma unroll
    for (int k = 0; k < 3; ++k) {
        wi1A[k].x = Wi1[k * 4 + 0]; wi1A[k].y = Wi1[k * 4 + 1];
        wi1B[k].x = Wi1[k * 4 + 2]; wi1B[k].y = Wi1[k * 4 + 3];
    }
    #pragma unroll
    for (int k = 0; k < 4; ++k) {
        wh1A[k].x = Wh1[k * 4 + 0]; wh1A[k].y = Wh1[k * 4 + 1];
        wh1B[k].x = Wh1[k * 4 + 2]; wh1B[k].y = Wh1[k * 4 + 3];
    }
    #pragma unroll
    for (int k = 0; k < 2; ++k) {
        wi2A[k].x = Wi2[k * 4 + 0]; wi2A[k].y = Wi2[k * 4 + 1];
        wi2B[k].x = Wi2[k * 4 + 2]; wi2B[k].y = Wi2[k * 4 + 3];
    }
    #pragma unroll
    for (int k = 0; k < 4; ++k) {
        wh2A[k].x = Wh2[k * 4 + 0]; wh2A[k].y = Wh2[k * 4 + 1];
        wh2B[k].x = Wh2[k * 4 + 2]; wh2B[k].y = Wh2[k * 4 + 3];
    }
    b1A.x = b1[0]; b1A.y = b1[1]; b1B.x = b1[2]; b1B.y = b1[3];
    b2A.x = b2[0]; b2A.y = b2[1]; b2B.x = b2[2]; b2B.y = b2[3];

    float s0 = 0.f, s1 = 0.f, s2 = 0.f, s3 = 0.f;

    // Phase 1: inputs (y[l,t], h[l,t+1,0:2]), t = 0..NY-1
    for (int t = 0; t < NY_; ++t) {
        float  in0 = yT[(size_t)t * L_SEQ + l];
        float2 hp  = h2[(size_t)(t + 1) * L_SEQ + l];
        __builtin_prefetch(&h2[(size_t)(t + 17) * L_SEQ + l], 0, 1);
        // input-only part: independent of recurrence, overlaps previous tanh
        v2f baseA = b1A + in0 * wi1A[0] + hp.x * wi1A[1] + hp.y * wi1A[2];
        v2f baseB = b1B + in0 * wi1B[0] + hp.x * wi1B[1] + hp.y * wi1B[2];
        // recurrent part: shallow tree, packed
        v2f accA = (s0 * wh1A[0] + s1 * wh1A[1]) + (s2 * wh1A[2] + s3 * wh1A[3]);
        v2f accB = (s0 * wh1B[0] + s1 * wh1B[1]) + (s2 * wh1B[2] + s3 * wh1B[3]);
        v2f nA = baseA + accA;
        v2f nB = baseB + accB;
        tanh4(nA.x, nA.y, nB.x, nB.y, s0, s1, s2, s3);
        sT[(size_t)t * L_SEQ + l] = make_float4(s0, s1, s2, s3);
    }
    // Phase 2: inputs h[l, NY+1+t, 0:2], t = 0..T-NY-2
    for (int t = 0; t < T_LEN - NY_ - 1; ++t) {
        float2 hp = h2[(size_t)(NY_ + 1 + t) * L_SEQ + l];
        __builtin_prefetch(&h2[(size_t)(NY_ + 17 + t) * L_SEQ + l], 0, 1);
        v2f baseA = b2A + hp.x * wi2A[0] + hp.y * wi2A[1];
        v2f baseB = b2B + hp.x * wi2B[0] + hp.y * wi2B[1];
        v2f accA = (s0 * wh2A[0] + s1 * wh2A[1]) + (s2 * wh2A[2] + s3 * wh2A[3]);
        v2f accB = (s0 * wh2B[0] + s1 * wh2B[1]) + (s2 * wh2B[2] + s3 * wh2B[3]);
        v2f nA = baseA + accA;
        v2f nB = baseB + accB;
        tanh4(nA.x, nA.y, nB.x, nB.y, s0, s1, s2, s3);
        sT[(size_t)(NY_ + t) * L_SEQ + l] = make_float4(s0, s1, s2, s3);
    }
}

// ---------------------------------------------------------------------------
// 4) Output head. One wave per 16-row tile (16 consecutive t of one sequence).
//    First GEMM (K=4) = single V_WMMA_F32_16X16X4_F32.
//    A 16x4 f32 layout (ISA 7.12.2): lanes 0-15 hold K=0,1 of row M=lane,
//    lanes 16-31 hold K=2,3 of row M=lane-16. B 4x16 mirrored:
//    VGPR0 = rows K=0 (lanes 0-15) / K=2 (lanes 16-31), VGPR1 = K=1 / K=3,
//    column N = lane%16. C/D: lane holds column N=lane%16, VGPR v = row
//    v + (lane<16 ? 0 : 8).
// ---------------------------------------------------------------------------
__global__ __launch_bounds__(256) void head_kernel(
    const float4* __restrict__ sT,
    const float* __restrict__ Wd1, const float* __restrict__ bd1,
    const float* __restrict__ Wd2, const float* __restrict__ bd2,
    float* __restrict__ out)
{
    __shared__ float rt[8][16 * 16];
    const int lane = threadIdx.x & 31;
    const int wave = threadIdx.x >> 5;
    const int tile = blockIdx.x * 8 + wave;      // 256 tiles per sequence
    const int l    = tile >> 8;
    const int t0   = (tile & 255) << 4;
    const int m    = lane & 15;
    const int hi   = lane >> 4;
    const int t    = t0 + m;

    float4 sv = make_float4(0.f, 0.f, 0.f, 0.f);
    if (t < T_OUT) sv = sT[(size_t)t * L_SEQ + l];

    v2f a; a.x = hi ? sv.z : sv.x; a.y = hi ? sv.w : sv.y;

    v2f b; b.x = 0.f; b.y = 0.f;
    if (m < D1_) {
        b.x = Wd1[(hi ? 2 : 0) * D1_ + m];
        b.y = Wd1[(hi ? 3 : 1) * D1_ + m];
    }

    v8f c = {};
    c = __builtin_amdgcn_wmma_f32_16x16x4_f32(false, a, false, b, (short)0, c, false, false);

    const float bias = (m < D1_) ? bd1[m] : 0.f;
    #pragma unroll
    for (int v = 0; v < 8; ++v) {
        float r = c[v] + bias;
        r = r > 0.f ? r : 0.f;
        rt[wave][(v + (hi ? 8 : 0)) * 16 + m] = r;
    }
    __syncthreads();

    if (lane < 16 && (t0 + lane) < T_OUT) {
        const float* rr = &rt[wave][lane * 16];
        float o0 = bd2[0], o1 = bd2[1];
        #pragma unroll
        for (int j = 0; j < D1_; ++j) {
            o0 += rr[j] * Wd2[j * 2 + 0];
            o1 += rr[j] * Wd2[j * 2 + 1];
        }
        size_t oi = ((size_t)l * T_OUT + (t0 + lane)) * 2;
        out[oi]     = o0;
        out[oi + 1] = o1;
    }
}

extern "C" void kernel_launch(void* const* d_in, const int* in_sizes, int n_in,
                              void* d_out, int out_size, void* d_ws, size_t ws_size,
                              hipStream_t stream) {
    (void)in_sizes; (void)n_in; (void)out_size; (void)ws_size;
    const float* x     = (const float*)d_in[0];
    const float* y     = (const float*)d_in[1];
    const float* embed = (const float*)d_in[2];
    const float* Wp1   = (const float*)d_in[3];
    const float* bp1   = (const float*)d_in[4];
    const float* Wp2   = (const float*)d_in[5];
    const float* bp2   = (const float*)d_in[6];
    const float* Wi1   = (const float*)d_in[7];
    const float* Wh1   = (const float*)d_in[8];
    const float* b1    = (const float*)d_in[9];
    const float* Wi2   = (const float*)d_in[10];
    const float* Wh2   = (const float*)d_in[11];
    const float* b2    = (const float*)d_in[12];
    const float* Wd1   = (const float*)d_in[13];
    const float* bd1   = (const float*)d_in[14];
    const float* Wd2   = (const float*)d_in[15];
    const float* bd2   = (const float*)d_in[16];

    char* ws = (char*)d_ws;
    float2* h2 = (float2*)(ws);                    // [T][L]   float2 : 33,554,432 B
    float*  yT = (float*)(ws + 33554432);          // [NY][L]  float  :  8,388,608 B
    float4* sT = (float4*)(ws + 41943040);         // [T-1][L] float4 : 67,092,480 B

    preprocess_kernel<<<dim3(T_LEN / 64, L_SEQ / 16), 256, 0, stream>>>(
        x, embed, Wp1, bp1, Wp2, bp2, h2);
    ytranspose_kernel<<<dim3(NY_ / 32, L_SEQ / 32), 256, 0, stream>>>(y, yT);
    rnn_kernel<<<L_SEQ / 32, 32, 0, stream>>>(yT, h2, Wi1, Wh1, b1, Wi2, Wh2, b2, sT);
    head_kernel<<<(L_SEQ * 256) / 8, 256, 0, stream>>>(sT, Wd1, bd1, Wd2, bd2, (float*)d_out);
}